// ChamferLoss_21801253994783
// MI455X (gfx1250) — compile-verified
//
#include <hip/hip_runtime.h>
#include <hip/hip_bf16.h>

typedef __attribute__((ext_vector_type(2))) float v2f;
typedef __attribute__((ext_vector_type(8))) float v8f;

#define B_  8
#define N_  4096
#define M_  4096
#define ROWTILES   (N_ / 16)        // 256
#define COLTILES   (M_ / 16)        // 256
#define COLCHUNKS  8                // each wave handles COLTILES/COLCHUNKS = 32 column tiles
#define TILES_PER_CHUNK (COLTILES / COLCHUNKS)
#define WAVES_TOTAL (B_ * ROWTILES * COLCHUNKS)   // 16384
#define WAVES_PER_BLOCK 8
#define BLOCKS_MAIN (WAVES_TOTAL / WAVES_PER_BLOCK) // 2048

#define INF_BITS 0x7F800000u

// ---------------------------------------------------------------------------
// Fill dist1/dist2 (as uint bit patterns) with +inf
// ---------------------------------------------------------------------------
__global__ void chamfer_init_kernel(unsigned int* buf, int n) {
    int i = blockIdx.x * blockDim.x + threadIdx.x;
    if (i < n) buf[i] = INF_BITS;
}

// ---------------------------------------------------------------------------
// Main kernel: one wave = (batch, row tile of 16 pred points, chunk of 32
// column tiles). Each tile's 16x16 squared-distance block is produced by a
// single V_WMMA_F32_16X16X4_F32:
//   A(m,:) = [-2px,-2py,-2pz,||p||^2],  B(:,n) = [tx,ty,tz,1],  C(m,n)=||t||^2
// Clamp max(d,0) is done as signed-int max on the bit pattern (exact for
// non-NaN IEEE floats); all min-reductions run as unsigned-int mins (valid
// for nonnegative floats). No float canonicalization ops needed.
// ---------------------------------------------------------------------------
__global__ __launch_bounds__(256) void chamfer_wmma_kernel(
    const float* __restrict__ pred, const float* __restrict__ target,
    unsigned int* __restrict__ dist1u, unsigned int* __restrict__ dist2u) {

    const int lane   = threadIdx.x & 31;
    const int waveIB = threadIdx.x >> 5;
    const int wid    = blockIdx.x * WAVES_PER_BLOCK + waveIB;   // [0, WAVES_TOTAL)

    const int b       = wid / (ROWTILES * COLCHUNKS);
    const int rem     = wid % (ROWTILES * COLCHUNKS);
    const int rowtile = rem / COLCHUNKS;
    const int chunk   = rem % COLCHUNKS;

    const int col16 = lane & 15;      // position within 16-wide group
    const int half  = lane >> 4;      // 0: K=0,1 / rows 0-7 ; 1: K=2,3 / rows 8-15

    // --- Load A tile: 16 pred points, row m = rowtile*16 + col16 -----------
    const int m      = rowtile * 16 + col16;
    const size_t pb  = ((size_t)b * N_ + m) * 3;
    const float px = pred[pb + 0];
    const float py = pred[pb + 1];
    const float pz = pred[pb + 2];
    const float p2 = px * px + py * py + pz * pz;

    // A layout (32-bit A 16x4): lanes 0-15 hold K=0,1 ; lanes 16-31 hold K=2,3
    v2f a;
    a.x = half ? (-2.0f * pz) : (-2.0f * px);
    a.y = half ? p2           : (-2.0f * py);

    // Running row-wise minima as uint bit patterns
    unsigned int rmin[8];
    #pragma unroll
    for (int k = 0; k < 8; ++k) rmin[k] = INF_BITS;

    const int t0 = chunk * TILES_PER_CHUNK;
    for (int t = t0; t < t0 + TILES_PER_CHUNK; ++t) {
        // --- Load B tile + C: target point for column n --------------------
        const int n     = t * 16 + col16;
        const size_t tb = ((size_t)b * M_ + n) * 3;
        const float tx = target[tb + 0];
        const float ty = target[tb + 1];
        const float tz = target[tb + 2];
        const float t2 = tx * tx + ty * ty + tz * tz;

        // B layout mirrors A: lanes 0-15 hold K=0,1 ; lanes 16-31 hold K=2,3
        v2f bv;
        bv.x = half ? tz   : tx;
        bv.y = half ? 1.0f : ty;

        v8f c;
        #pragma unroll
        for (int k = 0; k < 8; ++k) c[k] = t2;

        // --- One WMMA = full 16x16 squared-distance tile -------------------
        v8f d = __builtin_amdgcn_wmma_f32_16x16x4_f32(
            /*neg_a=*/false, a, /*neg_b=*/false, bv,
            /*c_mod=*/(short)0, c, /*reuse_a=*/false, /*reuse_b=*/false);

        // Clamp via signed-int max (== fmaxf(x,0) for non-NaN), then uint mins
        unsigned int u[8];
        #pragma unroll
        for (int k = 0; k < 8; ++k) {
            u[k] = (unsigned int)max((int)__float_as_uint(d[k]), 0);
            rmin[k] = min(rmin[k], u[k]);
        }
        unsigned int cmin = u[0];
        #pragma unroll
        for (int k = 1; k < 8; ++k) cmin = min(cmin, u[k]);

        // Column min: combine the two 8-row halves (lane L <-> L+16)
        cmin = min(cmin, (unsigned int)__shfl_xor((int)cmin, 16, 32));
        if (lane < 16) {
            atomicMin(&dist2u[(size_t)b * M_ + t * 16 + lane], cmin);
        }
    }

    // --- Row minima: reduce across the 16 columns held by each half --------
    const size_t d1base = (size_t)b * N_ + rowtile * 16;
    #pragma unroll
    for (int r = 0; r < 8; ++r) {
        unsigned int v = rmin[r];
        v = min(v, (unsigned int)__shfl_xor((int)v, 1, 32));
        v = min(v, (unsigned int)__shfl_xor((int)v, 2, 32));
        v = min(v, (unsigned int)__shfl_xor((int)v, 4, 32));
        v = min(v, (unsigned int)__shfl_xor((int)v, 8, 32));
        if (col16 == 0) {   // lanes 0 and 16: rows r and r+8
            atomicMin(&dist1u[d1base + r + 8 * half], v);
        }
    }
}

// ---------------------------------------------------------------------------
// Deterministic single-block reduction: loss = (sum1 + sum2) / (2 * B * N)
// ---------------------------------------------------------------------------
__global__ void chamfer_reduce_kernel(const float* __restrict__ vals,
                                      float* __restrict__ out, int n) {
    __shared__ float sm[256];
    float s = 0.0f;
    for (int i = threadIdx.x; i < n; i += 256) s += vals[i];
    sm[threadIdx.x] = s;
    __syncthreads();
    for (int off = 128; off > 0; off >>= 1) {
        if ((int)threadIdx.x < off) sm[threadIdx.x] += sm[threadIdx.x + off];
        __syncthreads();
    }
    if (threadIdx.x == 0) out[0] = sm[0] * (1.0f / (float)(2 * B_ * N_));
}

extern "C" void kernel_launch(void* const* d_in, const int* in_sizes, int n_in,
                              void* d_out, int out_size, void* d_ws, size_t ws_size,
                              hipStream_t stream) {
    const float* pred   = (const float*)d_in[0];
    const float* target = (const float*)d_in[1];
    float* out = (float*)d_out;

    // d_ws layout: dist1 [B*N floats] | dist2 [B*M floats]
    unsigned int* dist1u = (unsigned int*)d_ws;
    unsigned int* dist2u = dist1u + (size_t)B_ * N_;
    const int ntot = B_ * N_ + B_ * M_;   // 65536

    chamfer_init_kernel<<<(ntot + 255) / 256, 256, 0, stream>>>(dist1u, ntot);

    chamfer_wmma_kernel<<<BLOCKS_MAIN, 32 * WAVES_PER_BLOCK, 0, stream>>>(
        pred, target, dist1u, dist2u);

    chamfer_reduce_kernel<<<1, 256, 0, stream>>>((const float*)d_ws, out, ntot);
}